// Cell_61856118996994
// MI455X (gfx1250) — compile-verified
//
#include <hip/hip_runtime.h>
#include <math.h>

#define N_NODES   50000
#define N_ADJ     6
#define N_EDGES   800000
#define N_HID_PREV 256
#define N_HID     128

typedef float v2f __attribute__((ext_vector_type(2)));
typedef float v8f __attribute__((ext_vector_type(8)));

// ---------------------------------------------------------------------------
// Zero-fill (float4 stores)
// ---------------------------------------------------------------------------
__global__ void zero_f4(float4* __restrict__ p, long long n4) {
    long long i = (long long)blockIdx.x * blockDim.x + threadIdx.x;
    if (i < n4) p[i] = make_float4(0.f, 0.f, 0.f, 0.f);
}

// ---------------------------------------------------------------------------
// h0 = x @ W + b   via V_WMMA_F32_16X16X4_F32 (exact fp32 path)
// One wave computes one 16x16 tile of the [50000,128] output.
// Tiles: 3125 along M, 8 along N.  8 waves / 256-thread block.
// ---------------------------------------------------------------------------
__global__ void gemm_bias_wmma(const float* __restrict__ x,
                               const float* __restrict__ W,
                               const float* __restrict__ bias,
                               float* __restrict__ out) {
    int wave = (blockIdx.x * blockDim.x + threadIdx.x) >> 5;
    int lane = threadIdx.x & 31;
    int tn = wave & 7;          // 8 tiles across N=128
    int tm = wave >> 3;         // 3125 tiles across M=50000
    if (tm >= N_NODES / 16) return;
    int m0 = tm * 16, n0 = tn * 16;

    int lm = lane & 15;         // row (A) / col (B) within tile
    int kh = (lane >> 4) * 2;   // lane-half selects K pair {0,1} or {2,3}

    v8f c = {};
    const float* aRow = x + (size_t)(m0 + lm) * N_HID_PREV + kh;
    const float* bCol = W + (size_t)kh * N_HID + n0 + lm;
    for (int k0 = 0; k0 < N_HID_PREV; k0 += 4) {
        v2f a, b;
        a.x = aRow[k0];                       // A[m, k0+kh]
        a.y = aRow[k0 + 1];                   // A[m, k0+kh+1]
        b.x = bCol[(size_t)k0 * N_HID];       // B[k0+kh,   n]
        b.y = bCol[(size_t)(k0 + 1) * N_HID]; // B[k0+kh+1, n]
        c = __builtin_amdgcn_wmma_f32_16x16x4_f32(
                false, a, false, b, (short)0, c, false, false);
    }

    float bv = bias[n0 + lm];
    int rbase = m0 + ((lane >> 4) << 3);      // lanes 16-31 hold rows M=8..15
    #pragma unroll
    for (int i = 0; i < 8; ++i)
        out[(size_t)(rbase + i) * N_HID + n0 + lm] = c[i] + bv;
}

// ---------------------------------------------------------------------------
// out[row,:] += w * val[e] * h[col,:]   (scatter-add SpMM)
// One wave handles 32 edges: each lane loads ONE edge's metadata (coalesced,
// no 32x redundancy), then the wave iterates the batch via __shfl broadcast.
// Per edge: 32 lanes x float4 = full 128-float row gather (512B coalesced),
// 4 non-returning global_atomic_add_f32 per lane into the accumulator.
// mode 0: k = idx            (raw adjacency index)
// mode 1: k = CSTR[idx]      = {0,2,4}[idx]  = 2*idx
// mode 2: k = CSTR_LAST[idx] = {0,2,4,5}[idx]
// ---------------------------------------------------------------------------
__global__ void spmm_atomic(const int*   __restrict__ rows_base,
                            const int*   __restrict__ cols_base,
                            const float* __restrict__ vals_base,
                            const int*   __restrict__ idx_ptr, int idx_off, int mode,
                            const float* __restrict__ w_ptr,   int w_off,
                            const float* __restrict__ h_in,
                            float*       __restrict__ out_acc) {
    int wid  = (blockIdx.x * blockDim.x + threadIdx.x) >> 5;
    int lane = threadIdx.x & 31;
    long long ebase = (long long)wid * 32;
    if (ebase >= N_EDGES) return;

    int raw = idx_ptr[idx_off];
    int k;
    if (mode == 0)       k = raw;
    else if (mode == 1)  k = raw * 2;
    else                 k = (raw == 3) ? (N_ADJ - 1) : raw * 2;

    float w = w_ptr[w_off];

    // Each lane owns one edge of the 32-edge batch.
    size_t eo = (size_t)k * N_EDGES + (size_t)ebase + lane;
    int   my_row = rows_base[eo];
    int   my_col = cols_base[eo];
    float my_val = vals_base[eo] * w;

    int lf = lane * 4;  // this lane's 4-feature slice

    #pragma unroll 4
    for (int j = 0; j < 32; ++j) {
        int   row = __shfl(my_row, j, 32);
        int   col = __shfl(my_col, j, 32);
        float s   = __shfl(my_val, j, 32);

        const float4 hv = *(const float4*)(h_in + (size_t)col * N_HID + lf);
        float* dst = out_acc + (size_t)row * N_HID + lf;
        atomicAdd(dst + 0, s * hv.x);
        atomicAdd(dst + 1, s * hv.y);
        atomicAdd(dst + 2, s * hv.z);
        atomicAdd(dst + 3, s * hv.w);
    }
}

// ---------------------------------------------------------------------------
// LayerNorm (eps=1e-5, no affine) + exact GELU, one wave per 128-wide row.
// ---------------------------------------------------------------------------
__global__ void ln_gelu(const float* __restrict__ in, float* __restrict__ out) {
    int wave = (blockIdx.x * blockDim.x + threadIdx.x) >> 5;
    int lane = threadIdx.x & 31;
    if (wave >= N_NODES) return;

    const float4 v = *(const float4*)(in + (size_t)wave * N_HID + lane * 4);
    float s = v.x + v.y + v.z + v.w;
    float q = v.x * v.x + v.y * v.y + v.z * v.z + v.w * v.w;
    #pragma unroll
    for (int off = 16; off > 0; off >>= 1) {
        s += __shfl_xor(s, off, 32);
        q += __shfl_xor(q, off, 32);
    }
    float mu  = s * (1.0f / N_HID);
    float var = q * (1.0f / N_HID) - mu * mu;
    float rs  = rsqrtf(var + 1e-5f);

    float4 o;
    float t;
    t = (v.x - mu) * rs; o.x = 0.5f * t * (1.0f + erff(t * 0.70710678118654752f));
    t = (v.y - mu) * rs; o.y = 0.5f * t * (1.0f + erff(t * 0.70710678118654752f));
    t = (v.z - mu) * rs; o.z = 0.5f * t * (1.0f + erff(t * 0.70710678118654752f));
    t = (v.w - mu) * rs; o.w = 0.5f * t * (1.0f + erff(t * 0.70710678118654752f));
    *(float4*)(out + (size_t)wave * N_HID + lane * 4) = o;
}

// ---------------------------------------------------------------------------
extern "C" void kernel_launch(void* const* d_in, const int* in_sizes, int n_in,
                              void* d_out, int out_size, void* d_ws, size_t ws_size,
                              hipStream_t stream) {
    const float* x          = (const float*)d_in[0];
    const int*   adj_rows   = (const int*)  d_in[1];
    const int*   adj_cols   = (const int*)  d_in[2];
    const float* adj_vals   = (const float*)d_in[3];
    const float* W          = (const float*)d_in[4];
    const float* b          = (const float*)d_in[5];
    const float* ws_seq0    = (const float*)d_in[6];
    const float* ws_seq1    = (const float*)d_in[7];
    const float* ws_res0    = (const float*)d_in[8];
    const float* ws_res1    = (const float*)d_in[9];
    const int*   idxes_seq0 = (const int*)  d_in[10];
    const int*   idxes_seq1 = (const int*)  d_in[11];
    const int*   idxes_res0 = (const int*)  d_in[12];
    const int*   idxes_res1 = (const int*)  d_in[13];
    float* out = (float*)d_out;

    const size_t NF = (size_t)N_NODES * N_HID;   // 6.4M floats per state
    float* s0  = (float*)d_ws;
    float* s1  = s0 + NF;
    float* s2  = s1 + NF;
    float* s3  = s2 + NF;
    float* acc = s3 + NF;

    // Zero s1..s3 + acc (contiguous 4*NF floats); s0 is fully written by GEMM.
    {
        long long n4 = (long long)(4 * NF) / 4;
        int blocks = (int)((n4 + 255) / 256);
        zero_f4<<<blocks, 256, 0, stream>>>((float4*)s1, n4);
    }

    // s0 = x@W + b  (25000 waves = 3125 blocks of 8 waves)
    gemm_bias_wmma<<<(N_NODES / 16 * 8 + 7) / 8, 256, 0, stream>>>(x, W, b, s0);

    // 25000 waves (32 edges each) = 3125 blocks of 8 waves
    const int SPMM_BLOCKS = (N_EDGES / 32 + 7) / 8;

    // states[1] = op(idxes_seq0[0], ws_seq0[0], s0)
    spmm_atomic<<<SPMM_BLOCKS, 256, 0, stream>>>(adj_rows, adj_cols, adj_vals,
        idxes_seq0, 0, 0, ws_seq0, 0, s0, s1);

    // states[2] = op(idxes_seq0[1], ws_seq0[1], s1) + op(idxes_res0[0], ws_res0[0], s0)
    spmm_atomic<<<SPMM_BLOCKS, 256, 0, stream>>>(adj_rows, adj_cols, adj_vals,
        idxes_seq0, 1, 0, ws_seq0, 1, s1, s2);
    spmm_atomic<<<SPMM_BLOCKS, 256, 0, stream>>>(adj_rows, adj_cols, adj_vals,
        idxes_res0, 0, 0, ws_res0, 0, s0, s2);

    // states[3] = op(idxes_seq0[2], ws_seq0[2], s2)
    //           + op(idxes_res0[1], ws_res0[1], s0) + op(idxes_res0[2], ws_res0[2], s1)
    spmm_atomic<<<SPMM_BLOCKS, 256, 0, stream>>>(adj_rows, adj_cols, adj_vals,
        idxes_seq0, 2, 0, ws_seq0, 2, s2, s3);
    spmm_atomic<<<SPMM_BLOCKS, 256, 0, stream>>>(adj_rows, adj_cols, adj_vals,
        idxes_res0, 1, 0, ws_res0, 1, s0, s3);
    spmm_atomic<<<SPMM_BLOCKS, 256, 0, stream>>>(adj_rows, adj_cols, adj_vals,
        idxes_res0, 2, 0, ws_res0, 2, s1, s3);

    // acc = op(CSTR[idxes_seq1[0]], ws_seq1[0], s3)  (mode 1)
    spmm_atomic<<<SPMM_BLOCKS, 256, 0, stream>>>(adj_rows, adj_cols, adj_vals,
        idxes_seq1, 0, 1, ws_seq1, 0, s3, acc);

    // acc += op(CSTR_LAST[idxes_res1[i]], ws_res1[i], states[i])  (mode 2)
    spmm_atomic<<<SPMM_BLOCKS, 256, 0, stream>>>(adj_rows, adj_cols, adj_vals,
        idxes_res1, 0, 2, ws_res1, 0, s0, acc);
    spmm_atomic<<<SPMM_BLOCKS, 256, 0, stream>>>(adj_rows, adj_cols, adj_vals,
        idxes_res1, 1, 2, ws_res1, 1, s1, acc);
    spmm_atomic<<<SPMM_BLOCKS, 256, 0, stream>>>(adj_rows, adj_cols, adj_vals,
        idxes_res1, 2, 2, ws_res1, 2, s2, acc);

    // LayerNorm + exact GELU -> d_out   (50000 waves = 6250 blocks of 8 waves)
    ln_gelu<<<(N_NODES + 7) / 8, 256, 0, stream>>>(acc, out);
}